// TextEncoder_without_know_15642270892364
// MI455X (gfx1250) — compile-verified
//
#include <hip/hip_runtime.h>
#include <math.h>

// ---------------------------------------------------------------------------
// Problem dims (fixed by the reference)
// ---------------------------------------------------------------------------
#define N_BATCH 64
#define L_TOK   2050
#define D_IN    768
#define L1      512
#define OUT     300
#define OUT_PAD 304          // 19 tiles of 16
#define N_TILES 19
#define LDA     772          // LDS row stride for A (768 + 4 pad -> conflict-free)

typedef __attribute__((ext_vector_type(2))) float v2f;
typedef __attribute__((ext_vector_type(8))) float v8f;

// ---------------------------------------------------------------------------
// Kernel 0: zero-pad W_lin_ (768x300) -> Wp (768x304) so GEMM B loads are
// branch-free. 0.93MB, stays hot in the 192MB L2 across all GEMM blocks.
// ---------------------------------------------------------------------------
__global__ __launch_bounds__(256) void pad_w_kernel(const float* __restrict__ W,
                                                    float* __restrict__ Wp) {
    int i = blockIdx.x * 256 + threadIdx.x;           // over 768*304
    if (i >= D_IN * OUT_PAD) return;
    int k = i / OUT_PAD, n = i - k * OUT_PAD;
    Wp[i] = (n < OUT) ? W[k * OUT + n] : 0.0f;
}

// ---------------------------------------------------------------------------
// Kernel 1: fused span-mean-pool (global -> LDS) + f32 WMMA GEMM.
// One block = 16 consecutive spans of one batch. 2048 blocks x 256 threads.
// Emits v_wmma_f32_16x16x4_f32 (full f32 precision; GEMM is compute-trivial,
// the kernel is HBM-bound on reading t1 once = 400MB).
// Pre-LN x (stride OUT, guarded) is staged in out2 (second half of d_out),
// which is later overwritten by the softmax broadcast.
// ---------------------------------------------------------------------------
__global__ __launch_bounds__(256) void pool_gemm_kernel(
    const float* __restrict__ t1,
    const int*   __restrict__ starts,
    const int*   __restrict__ ends,
    const float* __restrict__ Wp,
    float*       __restrict__ xpre)   // [32768 x OUT] staging (in out2)
{
    __shared__ float As[16 * LDA];

    const int b   = blockIdx.x;           // 0..2047
    const int n   = b >> 5;               // batch   (32 groups per batch)
    const int j0  = (b & 31) << 4;        // first span of this group
    const int tid = threadIdx.x;

    // ---- Phase A: pool 16 spans (len 1..4 contiguous token rows) into LDS
    {
        const int r  = tid >> 4;          // local span row 0..15
        const int c0 = tid & 15;          // col phase: threads 0-15 coalesce 64B
        const int j  = j0 + r;
        const int s  = starts[n * L1 + j];
        const int e  = ends[n * L1 + j];
        const float inv = 1.0f / (float)(e - s);
        const float* base = t1 + ((size_t)n * L_TOK + 1) * D_IN;  // tok = t1[:,1:-1]
        for (int c = c0; c < D_IN; c += 16) {
            float acc = 0.0f;
            for (int t = s; t < e; ++t)
                acc += base[(size_t)t * D_IN + c];
            As[r * LDA + c] = acc * inv;
        }
    }
    __syncthreads();

    // ---- Phase B: WMMA. 8 waves round-robin over the 19 N-tiles.
    const int lane = tid & 31;
    const int wave = tid >> 5;
    const int half = lane >> 4;           // 0: lanes 0-15, 1: lanes 16-31
    const int nIdx = lane & 15;           // N within tile / M row for A frag
    const float* arow = &As[nIdx * LDA];  // A frag: M = lane%16

    for (int tile = wave; tile < N_TILES; tile += 8) {   // wave-uniform cond
        const int   col  = tile * 16 + nIdx;
        const float* bcol = Wp + col;
        v8f acc = {};
        for (int k0 = 0; k0 < D_IN; k0 += 4) {
            // ISA 16x4 f32 A layout: VGPR0 = K{0|2}, VGPR1 = K{1|3} per half
            const int ka = k0 + 2 * half;
            v2f a;  a.x = arow[ka];
                    a.y = arow[ka + 1];
            // B 4x16 f32: rows striped across lanes, mirrored K ordering
            v2f bb; bb.x = bcol[(size_t)ka * OUT_PAD];
                    bb.y = bcol[(size_t)(ka + 1) * OUT_PAD];
            acc = __builtin_amdgcn_wmma_f32_16x16x4_f32(
                      false, a, false, bb, (short)0, acc, false, false);
        }
        // D layout: VGPR v -> M = v + 8*half, N = lane%16
        const size_t rowbase = (size_t)b * 16;            // = n*512 + j0
        if (col < OUT) {
            for (int v = 0; v < 8; ++v) {
                const int mrow = v + 8 * half;
                xpre[(rowbase + mrow) * OUT + col] = acc[v];
            }
        }
    }
}

// ---------------------------------------------------------------------------
// Kernel 2: bias + LayerNorm + score dot. One wave32 per span row (8 rows per
// 256-thread block). Shuffle reductions, no LDS.
// ---------------------------------------------------------------------------
__global__ __launch_bounds__(256) void ln_score_kernel(
    const float* __restrict__ xpre,      // [32768 x OUT] (in out2)
    const float* __restrict__ bias,      // b_lin_ [300]
    const float* __restrict__ gamma,
    const float* __restrict__ beta,
    const float* __restrict__ Wl,        // W_lin [300]
    const float* __restrict__ bl,        // b_lin [1]
    float*       __restrict__ xout,      // first half of d_out
    float*       __restrict__ score)     // [32768] in ws
{
    const int row  = blockIdx.x * 8 + (threadIdx.x >> 5);
    const int lane = threadIdx.x & 31;
    const float* xr = xpre + (size_t)row * OUT;

    float vals[10];
    float s = 0.0f, ss = 0.0f;
    int cnt = 0;
    for (int c = lane; c < OUT; c += 32) {
        float v = xr[c] + bias[c];
        vals[cnt++] = v;
        s += v; ss += v * v;
    }
    for (int off = 16; off > 0; off >>= 1) {
        s  += __shfl_xor(s,  off, 32);
        ss += __shfl_xor(ss, off, 32);
    }
    const float mu   = s * (1.0f / OUT);
    const float var  = ss * (1.0f / OUT) - mu * mu;
    const float rstd = rsqrtf(var + 1e-5f);

    float sc = 0.0f;
    cnt = 0;
    for (int c = lane; c < OUT; c += 32) {
        float v = (vals[cnt++] - mu) * rstd * gamma[c] + beta[c];
        xout[(size_t)row * OUT + c] = v;
        sc += v * Wl[c];
    }
    for (int off = 16; off > 0; off >>= 1) sc += __shfl_xor(sc, off, 32);
    if (lane == 0) score[row] = sc + bl[0];
}

// ---------------------------------------------------------------------------
// Kernel 3: masked softmax over the 512 spans of one batch + broadcast p to
// the (512,300) second output. One block per batch, 512 threads (16 waves).
// ---------------------------------------------------------------------------
__global__ __launch_bounds__(512) void softmax_bcast_kernel(
    const float*         __restrict__ score,
    const unsigned char* __restrict__ mask,   // jnp bool -> 1 byte
    const int*           __restrict__ lam,    // scalar
    float*               __restrict__ out2)   // second half of d_out
{
    __shared__ float red[16];
    __shared__ float p[L1];

    const int nb  = blockIdx.x;
    const int tid = threadIdx.x;
    const float lamf = (float)lam[0];

    float sc = score[nb * L1 + tid] * lamf;
    if (mask[nb * L1 + tid]) sc = -INFINITY;

    // --- max reduce (wave shuffles + LDS across 16 waves)
    float m = sc;
    for (int off = 16; off > 0; off >>= 1) m = fmaxf(m, __shfl_xor(m, off, 32));
    if ((tid & 31) == 0) red[tid >> 5] = m;
    __syncthreads();
    if (tid < 16) {
        float t = red[tid];
        for (int off = 8; off > 0; off >>= 1) t = fmaxf(t, __shfl_xor(t, off, 16));
        red[tid] = t;
    }
    __syncthreads();
    m = red[0];
    __syncthreads();

    // --- exp + sum reduce
    float e = expf(sc - m);                   // exp(-inf)=0 for masked
    float su = e;
    for (int off = 16; off > 0; off >>= 1) su += __shfl_xor(su, off, 32);
    if ((tid & 31) == 0) red[tid >> 5] = su;
    __syncthreads();
    if (tid < 16) {
        float t = red[tid];
        for (int off = 8; off > 0; off >>= 1) t += __shfl_xor(t, off, 16);
        red[tid] = t;
    }
    __syncthreads();
    p[tid] = e / red[0];
    __syncthreads();

    // --- broadcast p over (512,300)
    float* dst = out2 + (size_t)nb * L1 * OUT;
    for (int idx = tid; idx < L1 * OUT; idx += 512)
        dst[idx] = p[idx / OUT];
}

// ---------------------------------------------------------------------------
// Launch
// ---------------------------------------------------------------------------
extern "C" void kernel_launch(void* const* d_in, const int* in_sizes, int n_in,
                              void* d_out, int out_size, void* d_ws, size_t ws_size,
                              hipStream_t stream) {
    const float*         t1     = (const float*)d_in[0];
    const int*           starts = (const int*)d_in[1];
    const int*           ends   = (const int*)d_in[2];
    const unsigned char* mask   = (const unsigned char*)d_in[3];
    const float*         W_lin_ = (const float*)d_in[4];
    const float*         b_lin_ = (const float*)d_in[5];
    const float*         gamma  = (const float*)d_in[6];
    const float*         beta   = (const float*)d_in[7];
    const float*         W_lin  = (const float*)d_in[8];
    const float*         b_lin  = (const float*)d_in[9];
    const int*           lam    = (const int*)d_in[10];

    float* out1 = (float*)d_out;                               // x       (32768x300)
    float* out2 = out1 + (size_t)N_BATCH * L1 * OUT;           // score_out / xpre staging

    float* Wp    = (float*)d_ws;                               // 768*304 f32
    float* score = Wp + (size_t)D_IN * OUT_PAD;                // 32768   f32

    pad_w_kernel<<<(D_IN * OUT_PAD + 255) / 256, 256, 0, stream>>>(W_lin_, Wp);

    pool_gemm_kernel<<<N_BATCH * (L1 / 16), 256, 0, stream>>>(
        t1, starts, ends, Wp, /*xpre=*/out2);

    ln_score_kernel<<<(N_BATCH * L1) / 8, 256, 0, stream>>>(
        out2, b_lin_, gamma, beta, W_lin, b_lin, out1, score);

    softmax_bcast_kernel<<<N_BATCH, 512, 0, stream>>>(score, mask, lam, out2);
}